// BasicQNN_27273042330353
// MI455X (gfx1250) — compile-verified
//
#include <hip/hip_runtime.h>

typedef __attribute__((ext_vector_type(2))) float v2f;
typedef __attribute__((ext_vector_type(4))) float v4f;
typedef __attribute__((ext_vector_type(8))) float v8f;

#define NQ 4
#define NL 2
#define DIM 16

// ---------------------------------------------------------------------------
// Kernel 1 (one wave, runs once per launch): build the 16x16 real symmetric
// observable matrix  A = Re(U^H Z0 U)  where U is the weight-only part of the
// circuit. Thread t (t<16) simulates basis state |t> through the circuit to
// get column t of U; then A[:,t] is formed from column inner products.
// ---------------------------------------------------------------------------
__global__ void qnn_build_obs(const float* __restrict__ w, float* __restrict__ A16)
{
    const int t   = threadIdx.x;
    const int col = t & 15;

    float sr[DIM], si[DIM];
#pragma unroll
    for (int b = 0; b < DIM; ++b) { sr[b] = (b == col) ? 1.0f : 0.0f; si[b] = 0.0f; }

#pragma unroll
    for (int layer = 0; layer < NL; ++layer) {
#pragma unroll
        for (int q = 0; q < NQ; ++q) {
            const int st = 1 << (3 - q);           // wire 0 = MSB of flat index
            const float thx = w[layer * 12 + q * 3 + 0];
            const float thy = w[layer * 12 + q * 3 + 1];
            const float thz = w[layer * 12 + q * 3 + 2];
            const float cx = __cosf(0.5f * thx), sx = __sinf(0.5f * thx);
            const float cy = __cosf(0.5f * thy), sy = __sinf(0.5f * thy);
            const float cz = __cosf(0.5f * thz), sz = __sinf(0.5f * thz);
#pragma unroll
            for (int i = 0; i < DIM; ++i) {
                if (i & st) continue;
                const int j = i | st;
                float a0r = sr[i], a0i = si[i], a1r = sr[j], a1i = si[j];
                // RX: [[c, -i s], [-i s, c]]
                float b0r =  cx * a0r + sx * a1i;
                float b0i =  cx * a0i - sx * a1r;
                float b1r =  sx * a0i + cx * a1r;
                float b1i = -sx * a0r + cx * a1i;
                // RY: [[c, -s], [s, c]]
                float c0r = cy * b0r - sy * b1r;
                float c0i = cy * b0i - sy * b1i;
                float c1r = sy * b0r + cy * b1r;
                float c1i = sy * b0i + cy * b1i;
                // RZ: diag(e^{-i th/2}, e^{+i th/2})
                sr[i] =  cz * c0r + sz * c0i;
                si[i] =  cz * c0i - sz * c0r;
                sr[j] =  cz * c1r - sz * c1i;
                si[j] =  cz * c1i + sz * c1r;
            }
        }
        // CNOT ring: (0,1),(1,2),(2,3),(3,0)
        const int ctrl[4] = {0, 1, 2, 3};
        const int targ[4] = {1, 2, 3, 0};
#pragma unroll
        for (int e = 0; e < 4; ++e) {
            const int mc = 1 << (3 - ctrl[e]);
            const int mt = 1 << (3 - targ[e]);
#pragma unroll
            for (int i = 0; i < DIM; ++i) {
                if ((i & mc) && !(i & mt)) {
                    const int j = i | mt;
                    float tr = sr[i]; sr[i] = sr[j]; sr[j] = tr;
                    float ti = si[i]; si[i] = si[j]; si[j] = ti;
                }
            }
        }
    }

    __shared__ float Ur[DIM][DIM];
    __shared__ float Ui[DIM][DIM];
    if (t < 16) {
#pragma unroll
        for (int k = 0; k < DIM; ++k) { Ur[k][col] = sr[k]; Ui[k][col] = si[k]; }
    }
    __syncthreads();
    if (t < 16) {
#pragma unroll
        for (int m = 0; m < DIM; ++m) {
            float acc = 0.0f;
#pragma unroll
            for (int k = 0; k < DIM; ++k) {
                const float sgn = (k < 8) ? 1.0f : -1.0f;  // Z on wire 0 (MSB)
                acc += sgn * (Ur[k][m] * Ur[k][col] + Ui[k][m] * Ui[k][col]);
            }
            A16[m * DIM + col] = acc;   // A is symmetric; store row-major
        }
    }
}

// ---------------------------------------------------------------------------
// Kernel 2: batched quadratic form  out_s = psi_s^T A psi_s  via WMMA.
// One wave = 16 samples; lane l owns sample (l & 15), half-waves carry the
// K = {2*hi, 2*hi+1} striping of the 16x4 / 4x16 f32 WMMA operand layouts.
// psi is never materialized as an array: every needed element factors as
// t01[.] * t23[.], so only 4 scalar cndmask selects depend on `hi`.
// ---------------------------------------------------------------------------
__global__ void __launch_bounds__(256)
qnn_batch(const float* __restrict__ x, const float* __restrict__ A16,
          float* __restrict__ out, int B)
{
    const int lane = threadIdx.x & 31;
    const int wave = blockIdx.x * (blockDim.x >> 5) + (threadIdx.x >> 5);
    const int base = wave << 4;                 // 16 samples per wave
    if (base >= B) return;                      // wave-uniform: EXEC stays full
    const int m  = lane & 15;                   // sample-in-tile == matrix row
    const int hi = lane >> 4;                   // half-wave (K/M high selector)

    // --- angle encoding factors: psi[b] = t01[b>>2] * t23[b&3]
    const v4f xv = *(const v4f*)(x + (size_t)(base + m) * NQ);   // one B128 load
    const float c0 = __cosf(0.5f * xv.x), s0 = __sinf(0.5f * xv.x);
    const float c1 = __cosf(0.5f * xv.y), s1 = __sinf(0.5f * xv.y);
    const float c2 = __cosf(0.5f * xv.z), s2 = __sinf(0.5f * xv.z);
    const float c3 = __cosf(0.5f * xv.w), s3 = __sinf(0.5f * xv.w);
    const float t01_0 = c0 * c1, t01_1 = c0 * s1, t01_2 = s0 * c1, t01_3 = s0 * s1;
    const float t23_0 = c2 * c3, t23_1 = c2 * s3, t23_2 = s2 * c3, t23_3 = s2 * s3;

    // B-operand needs psi[4*kc + 2*hi + j] = t01[kc] * t23[2*hi + j]
    const float u0 = hi ? t23_2 : t23_0;
    const float u1 = hi ? t23_3 : t23_1;

    // --- D = A * Psi^T, K=16 as 4 chunks of 16x16x4 f32 WMMA
    // WMMA-A (16x4): lane=M=m, VGPR j holds K = kc*4 + j + 2*hi (float2 load)
    const float* arow = A16 + m * DIM + 2 * hi;
    const v2f a0 = *(const v2f*)(arow + 0);
    const v2f a1 = *(const v2f*)(arow + 4);
    const v2f a2 = *(const v2f*)(arow + 8);
    const v2f a3 = *(const v2f*)(arow + 12);
    // WMMA-B (4x16): lane=N (sample), VGPR j holds K = kc*4 + j + 2*hi
    v2f b0, b1, b2, b3;
    b0.x = t01_0 * u0;  b0.y = t01_0 * u1;
    b1.x = t01_1 * u0;  b1.y = t01_1 * u1;
    b2.x = t01_2 * u0;  b2.y = t01_2 * u1;
    b3.x = t01_3 * u0;  b3.y = t01_3 * u1;

    v8f acc = {};
    acc = __builtin_amdgcn_wmma_f32_16x16x4_f32(false, a0, false, b0, (short)0, acc, false, false);
    acc = __builtin_amdgcn_wmma_f32_16x16x4_f32(false, a1, false, b1, (short)0, acc, false, false);
    acc = __builtin_amdgcn_wmma_f32_16x16x4_f32(false, a2, false, b2, (short)0, acc, false, false);
    acc = __builtin_amdgcn_wmma_f32_16x16x4_f32(false, a3, false, b3, (short)0, acc, false, false);

    // --- out_n = sum_m psi_n[m] * D[m][n]; VGPR r holds D[r + 8*hi][n],
    //     psi[r + 8*hi] = t01[(r>>2) + 2*hi] * t23[r&3]
    const float w0 = hi ? t01_2 : t01_0;
    const float w1 = hi ? t01_3 : t01_1;
    float sum0 = t23_0 * acc[0];
    sum0 = fmaf(t23_1, acc[1], sum0);
    sum0 = fmaf(t23_2, acc[2], sum0);
    sum0 = fmaf(t23_3, acc[3], sum0);
    float sum1 = t23_0 * acc[4];
    sum1 = fmaf(t23_1, acc[5], sum1);
    sum1 = fmaf(t23_2, acc[6], sum1);
    sum1 = fmaf(t23_3, acc[7], sum1);
    float partial = fmaf(w0, sum0, w1 * sum1);

    partial += __shfl_xor(partial, 16, 32);     // combine the two half-waves
    if (hi == 0) out[base + m] = partial;       // 16 coalesced floats per wave
}

// ---------------------------------------------------------------------------
extern "C" void kernel_launch(void* const* d_in, const int* in_sizes, int n_in,
                              void* d_out, int out_size, void* d_ws, size_t ws_size,
                              hipStream_t stream)
{
    const float* x = (const float*)d_in[0];     // [B, 4] f32
    const float* w = (const float*)d_in[1];     // [24]   f32
    float* out = (float*)d_out;                 // [B]    f32
    float* A16 = (float*)d_ws;                  // 256 f32 scratch (observable)
    const int B = in_sizes[0] / NQ;

    qnn_build_obs<<<1, 32, 0, stream>>>(w, A16);

    const int wavesPerBlock = 8;                            // 256 threads
    const int samplesPerBlock = wavesPerBlock * 16;         // 128
    const int blocks = (B + samplesPerBlock - 1) / samplesPerBlock;
    qnn_batch<<<blocks, wavesPerBlock * 32, 0, stream>>>(x, A16, out, B);
}